// LIFAct_csr_t_9852654977541
// MI455X (gfx1250) — compile-verified
//
#include <hip/hip_runtime.h>

// LIF forward: out[t] = H(u_t - 1), u carried with decay 0.25 and hard reset.
// Shape [T=4, B=64, C=256, H=32, W=32] fp32, time-major. Purely elementwise in
// time => memory-bound streaming kernel. Roofline: 536.9 MB / 23.3 TB/s ~= 23 us;
// ~4 VALU ops per element-step is >50x below the compute ceiling. No matrix math
// => WMMA correctly unused; the CDNA5-rewarded path is b128 NT loads/stores.

typedef float v4f __attribute__((ext_vector_type(4)));

#define LIF_VTH   1.0f
#define LIF_DECAY 0.25f
#define LIF_T     4

__global__ __launch_bounds__(256) void lif_forward_kernel(
    const float* __restrict__ xg,   // [T, N] flattened
    float*       __restrict__ og,   // [T, N]
    unsigned n4)                    // float4 chunks per time slice (fits 32-bit)
{
    // 32-bit divergent index + uniform (scalar) slice bases => backend can use
    // saddr-form global_load_b128 (SGPR base + 32-bit VGPR offset): one
    // v_lshlrev_b32 shared by all 8 accesses instead of v_add_nc_u64 chains.
    unsigned i = blockIdx.x * 256u + threadIdx.x;
    if (i >= n4) return;

    const v4f* __restrict__ x0 = (const v4f*)xg;                 // uniform bases
    const v4f* __restrict__ x1 = x0 + (size_t)n4;
    const v4f* __restrict__ x2 = x0 + 2 * (size_t)n4;
    const v4f* __restrict__ x3 = x0 + 3 * (size_t)n4;
    v4f* __restrict__ o0 = (v4f*)og;
    v4f* __restrict__ o1 = o0 + (size_t)n4;
    v4f* __restrict__ o2 = o0 + 2 * (size_t)n4;
    v4f* __restrict__ o3 = o0 + 3 * (size_t)n4;

    // All four loads are address-independent: issue before the dependent chain
    // so 4x global_load_b128 (non-temporal) are in flight per lane.
    v4f a0 = __builtin_nontemporal_load(x0 + i);
    v4f a1 = __builtin_nontemporal_load(x1 + i);
    v4f a2 = __builtin_nontemporal_load(x2 + i);
    v4f a3 = __builtin_nontemporal_load(x3 + i);

    v4f u, s;

    // t = 0 peeled: u0 = 0 => u = x (avoids fma-with-zero: 4 dead v_add_f32).
    // (Only deviation is -0 vs +0 in u, which cannot affect any u>1 compare.)
#pragma unroll
    for (int c = 0; c < 4; ++c) {
        float uu  = a0[c];
        bool fire = uu > LIF_VTH;      // v_cmp -> VCC
        s[c] = fire ? 1.0f : 0.0f;     // v_cndmask (reuses VCC)
        u[c] = fire ? 0.0f : uu;       // v_cndmask (reuses VCC)
    }
    __builtin_nontemporal_store(s, o0 + i);

#pragma unroll
    for (int c = 0; c < 4; ++c) {
        float uu  = __builtin_fmaf(u[c], LIF_DECAY, a1[c]);
        bool fire = uu > LIF_VTH;
        s[c] = fire ? 1.0f : 0.0f;
        u[c] = fire ? 0.0f : uu;
    }
    __builtin_nontemporal_store(s, o1 + i);

#pragma unroll
    for (int c = 0; c < 4; ++c) {
        float uu  = __builtin_fmaf(u[c], LIF_DECAY, a2[c]);
        bool fire = uu > LIF_VTH;
        s[c] = fire ? 1.0f : 0.0f;
        u[c] = fire ? 0.0f : uu;
    }
    __builtin_nontemporal_store(s, o2 + i);

#pragma unroll
    for (int c = 0; c < 4; ++c) {
        float uu  = __builtin_fmaf(u[c], LIF_DECAY, a3[c]);
        bool fire = uu > LIF_VTH;
        s[c] = fire ? 1.0f : 0.0f;
        u[c] = fire ? 0.0f : uu;
    }
    __builtin_nontemporal_store(s, o3 + i);
}

extern "C" void kernel_launch(void* const* d_in, const int* in_sizes, int n_in,
                              void* d_out, int out_size, void* d_ws, size_t ws_size,
                              hipStream_t stream)
{
    const float* x   = (const float*)d_in[0];
    float*       out = (float*)d_out;

    const long long total = (long long)in_sizes[0]; // T*B*C*H*W = 67,108,864
    const long long N     = total / LIF_T;          // elements per time slice
    const unsigned  n4    = (unsigned)(N / 4);      // float4 chunks per slice

    const int block = 256;                          // 8 wave32s per block
    const unsigned grid = (n4 + block - 1) / block; // 16384 blocks

    lif_forward_kernel<<<grid, block, 0, stream>>>(x, out, n4);
}